// FALayer_28449863368913
// MI455X (gfx1250) — compile-verified
//
#include <hip/hip_runtime.h>
#include <hip/hip_bf16.h>

typedef __attribute__((ext_vector_type(2))) float v2f;
typedef __attribute__((ext_vector_type(8))) float v8f;

// ---------------------------------------------------------------------------
// Kernel 1: per-node projections p_dst = h @ gate_w[0,:D], p_src = h @ gate_w[0,D:]
// One wave32 per 16-row tile of h, accumulated with V_WMMA_F32_16X16X4_F32.
//
// A (16x4 f32) layout per ISA: lanes 0-15 -> M=0..15 holding K=(k0,k0+1);
//                              lanes 16-31 -> M=0..15 holding K=(k0+2,k0+3).
// B (4x16 f32): same dual-half striping; columns N=0 (w_dst) and N=1 (w_src)
//               carry the weights. Columns >=2 load arbitrary valid LDS data:
//               WMMA has no cross-column mixing, and we never read D[:,n>=2],
//               so no masking is needed (no exceptions are generated either).
// Rows >= N: address clamped to N-1; garbage only lands in output rows that
//               the `ro < N` store guard discards. No masking needed.
// C/D (16x16 f32): VGPR r, lanes 0-15 -> M=r, N=lane; lanes 16-31 -> M=r+8.
// => column 0 of D lives in lanes {0,16}, column 1 in lanes {1,17}.
// ---------------------------------------------------------------------------
__global__ void fa_node_project(const float* __restrict__ h,
                                const float* __restrict__ gw,   // [2*D]
                                float* __restrict__ p_dst,      // [N]
                                float* __restrict__ p_src,      // [N]
                                int N, int D) {
    // stage gate_w in LDS (2*D floats; D=128 -> 1 KB)
    __shared__ float sgw[512];
    for (int i = threadIdx.x; i < 2 * D; i += blockDim.x) sgw[i] = gw[i];
    __syncthreads();

    const int wave = (int)((blockIdx.x * blockDim.x + threadIdx.x) >> 5);
    const int lane = (int)(threadIdx.x & 31);
    const int r0 = wave * 16;
    if (r0 >= N) return;

    const int m  = lane & 15;   // row within tile (A) / column (B,C,D)
    const int kh = lane >> 4;   // which K-pair this half-wave holds

    // clamp address only; out-of-range rows feed dead output rows
    int row = r0 + m;
    row = (row < N) ? row : (N - 1);
    const float* hrow = h + (size_t)row * (size_t)D;

    // lanes m==0 -> w_dst, m==1 -> w_src; m>=2 -> harmless valid loads
    const float* bp = sgw + ((m & 1) ? D : 0);

    v8f c = {};
    #pragma unroll
    for (int k0 = 0; k0 < D; k0 += 4) {
        const int k = k0 + 2 * kh;                      // 8-byte aligned
        const v2f a = *(const v2f*)(hrow + k);          // global_load_b64
        const v2f b = *(const v2f*)(bp + k);            // ds_load_b64

        // 8 args: (neg_a, A, neg_b, B, c_mod, C, reuse_a, reuse_b)
        c = __builtin_amdgcn_wmma_f32_16x16x4_f32(false, a, false, b,
                                                  (short)0, c, false, false);
    }

    // lanes with column index 0 -> p_dst, column index 1 -> p_src
    if (m < 2) {
        float* pout = (m == 0) ? p_dst : p_src;
        #pragma unroll
        for (int r = 0; r < 8; ++r) {
            const int ro = r0 + r + 8 * kh;
            if (ro < N) pout[ro] = c[r];
        }
    }
}

// ---------------------------------------------------------------------------
// Kernel 2: per-edge coefficient + gather h[src] + atomic scatter-add into
// z[dst]. One wave32 per edge: 4 f32 channels per lane (global_load_b128 +
// 4x global_atomic_add_f32). h (51 MB) and z (51 MB) both fit in the 192 MB
// L2, so gathers and atomics are L2-resident; HBM sees only compulsory
// traffic. Coefficient operands are forced wave-uniform (readfirstlane) so
// they lower to the scalar/SMEM path and keep the VMEM pipe free for the
// 512 B payload + atomics.
// ---------------------------------------------------------------------------
__global__ void fa_edge_scatter(const float* __restrict__ h,
                                const float* __restrict__ dnorm,   // [N]
                                const float* __restrict__ yes_no,  // [E]
                                const float* __restrict__ gate_b,  // [1]
                                const float* __restrict__ yes_w,   // [1]
                                const float* __restrict__ no_w,    // [1]
                                const int*   __restrict__ src,     // [E]
                                const int*   __restrict__ dst,     // [E]
                                const float* __restrict__ p_dst,   // [N]
                                const float* __restrict__ p_src,   // [N]
                                float* __restrict__ z,             // [N,D]
                                int E, int D) {
    // edge index is identical across the wave: make it an SGPR
    const int e = __builtin_amdgcn_readfirstlane(
        (int)((blockIdx.x * blockDim.x + threadIdx.x) >> 5));
    const int lane = (int)(threadIdx.x & 31);
    if (e >= E) return;

    const int s = src[e];    // scalar loads (uniform index)
    const int t = dst[e];

    const float g   = tanhf(p_dst[t] + p_src[s] + gate_b[0]);
    const float yn0 = yes_no[e];
    const float yn  = tanhf(yn0 * yes_w[0] + (1.0f - yn0) * no_w[0]);
    const float w   = (g + yn) * 0.5f * dnorm[t] * dnorm[s];

    const float* hs = h + (size_t)s * (size_t)D;
    float*       zt = z + (size_t)t * (size_t)D;

    // D channels, 4 per lane per iteration (D=128 -> single iteration)
    for (int j = lane * 4; j < D; j += 128) {
        const float4 v = *(const float4*)(hs + j);      // global_load_b128
        atomicAdd(zt + j + 0, v.x * w);
        atomicAdd(zt + j + 1, v.y * w);
        atomicAdd(zt + j + 2, v.z * w);
        atomicAdd(zt + j + 3, v.w * w);
    }
}

extern "C" void kernel_launch(void* const* d_in, const int* in_sizes, int n_in,
                              void* d_out, int out_size, void* d_ws, size_t ws_size,
                              hipStream_t stream) {
    // setup_inputs() order:
    // 0: h [N,D] f32   1: d [N] f32      2: yes_no [E] f32
    // 3: gate_w [2D]   4: gate_b [1]     5: yes_weight [1]  6: no_weight [1]
    // 7: src [E] int   8: dst [E] int
    const float* h      = (const float*)d_in[0];
    const float* dnorm  = (const float*)d_in[1];
    const float* yes_no = (const float*)d_in[2];
    const float* gate_w = (const float*)d_in[3];
    const float* gate_b = (const float*)d_in[4];
    const float* yes_w  = (const float*)d_in[5];
    const float* no_w   = (const float*)d_in[6];
    const int*   src    = (const int*)d_in[7];
    const int*   dst    = (const int*)d_in[8];
    float* z = (float*)d_out;

    const int N = in_sizes[1];            // d has N elements
    const int E = in_sizes[2];            // yes_no has E elements
    const int D = in_sizes[0] / N;        // h is [N, D]

    // workspace: p_dst [N], p_src [N]
    float* p_dst = (float*)d_ws;
    float* p_src = p_dst + N;

    // z is an accumulator -> must start at zero every call (capture-safe memset)
    hipMemsetAsync(d_out, 0, (size_t)out_size * sizeof(float), stream);

    // projection: one wave per 16-row tile
    {
        const int tiles   = (N + 15) / 16;
        const int threads = 256;
        const int blocks  = (tiles * 32 + threads - 1) / threads;
        fa_node_project<<<blocks, threads, 0, stream>>>(h, gate_w, p_dst, p_src, N, D);
    }

    // edge scatter: one wave per edge
    {
        const int threads = 256;                       // 8 waves/block
        const long long total = (long long)E * 32;
        const int blocks = (int)((total + threads - 1) / threads);
        fa_edge_scatter<<<blocks, threads, 0, stream>>>(h, dnorm, yes_no, gate_b,
                                                        yes_w, no_w, src, dst,
                                                        p_dst, p_src, z, E, D);
    }
}